// TaskNearestAssigner_30099130810995
// MI455X (gfx1250) — compile-verified
//
#include <hip/hip_runtime.h>
#include <hip/hip_bf16.h>
#include <stdint.h>

// ---------------- problem constants (fixed by the reference) ----------------
#define NA_   3
#define NCLS_ 80
#define BS_   32
#define NMB_  128
#define GS_   80
#define NG_   (GS_ * GS_)          // 6400

typedef float        v4f   __attribute__((ext_vector_type(4)));
typedef unsigned int u32x4 __attribute__((ext_vector_type(4)));
typedef int          i32x4 __attribute__((ext_vector_type(4)));
typedef int          i32x8 __attribute__((ext_vector_type(8)));

#if __has_builtin(__builtin_amdgcn_tensor_load_to_lds) && __has_builtin(__builtin_amdgcn_s_wait_tensorcnt)
#define USE_TDM 1
#else
#define USE_TDM 0
#endif

// Stable comparator matching jax.lax.top_k tie-break (lower index wins on ties)
__device__ __forceinline__ bool kv_lt(float da, int ia, float db, int ib) {
    return (da < db) || ((da == db) && (ia < ib));
}

// Insert (d,i) into sorted top-3 (smallest first)
__device__ __forceinline__ void kv_insert(float d, int i,
                                          float& d0, int& i0,
                                          float& d1, int& i1,
                                          float& d2, int& i2) {
    if (kv_lt(d, i, d2, i2)) {
        if (kv_lt(d, i, d1, i1)) {
            d2 = d1; i2 = i1;
            if (kv_lt(d, i, d0, i0)) { d1 = d0; i1 = i0; d0 = d; i0 = i; }
            else                     { d1 = d;  i1 = i; }
        } else { d2 = d; i2 = i; }
    }
}

// ---------------- K1: top-3 nearest grid cells per gt (one wave32 per gt) ---
__global__ __launch_bounds__(256)
void k_topk(const float* __restrict__ gt_cxys, int* __restrict__ ws_topk) {
    const int wid  = (blockIdx.x * blockDim.x + threadIdx.x) >> 5;   // wave id == gt id
    const int lane = threadIdx.x & 31;
    if (wid >= BS_ * NMB_) return;

    const float cx = gt_cxys[wid * 2 + 0];
    const float cy = gt_cxys[wid * 2 + 1];

    float d0 = INFINITY, d1 = INFINITY, d2 = INFINITY;
    int   i0 = 0x7fffffff, i1 = 0x7fffffff, i2 = 0x7fffffff;

    for (int g = lane; g < NG_; g += 32) {
        const float gx = (float)(g % GS_) + 0.5f;
        const float gy = (float)(g / GS_) + 0.5f;
        const float d  = fabsf(gx - cx) + fabsf(gy - cy);
        kv_insert(d, g, d0, i0, d1, i1, d2, i2);
    }

    // wave32 butterfly merge of sorted triples
#pragma unroll
    for (int off = 16; off > 0; off >>= 1) {
        const float e0 = __shfl_xor(d0, off, 32);
        const int   j0 = __shfl_xor(i0, off, 32);
        const float e1 = __shfl_xor(d1, off, 32);
        const int   j1 = __shfl_xor(i1, off, 32);
        const float e2 = __shfl_xor(d2, off, 32);
        const int   j2 = __shfl_xor(i2, off, 32);
        kv_insert(e0, j0, d0, i0, d1, i1, d2, i2);
        kv_insert(e1, j1, d0, i0, d1, i1, d2, i2);
        kv_insert(e2, j2, d0, i0, d1, i1, d2, i2);
    }

    if (lane == 0) {
        ws_topk[wid * 3 + 0] = i0;
        ws_topk[wid * 3 + 1] = i1;
        ws_topk[wid * 3 + 2] = i2;
    }
}

// ---------------- K2: per-cell conflict resolution --------------------------
// One block per (batch, 256-cell chunk). gt centers staged into LDS via TDM.
__global__ __launch_bounds__(256)
void k_assign(const float* __restrict__ gt_cxys,
              const float* __restrict__ gt_cls,
              const float* __restrict__ mask_gt,
              const int*   __restrict__ ws_topk,
              int*         __restrict__ ws_packed) {
    __shared__ float s_cxy[NMB_ * 2];
    __shared__ int   s_k[NMB_ * 3];
    __shared__ int   s_m[NMB_];

    const int b     = blockIdx.x / (NG_ / 256);
    const int chunk = blockIdx.x % (NG_ / 256);
    const int g     = chunk * 256 + threadIdx.x;

#if USE_TDM
    // Tensor Data Mover: async DMA of 128 gt centers (256 f32 = 1KB) -> LDS.
    if (threadIdx.x < 32) {
        const uint64_t ga  = (uint64_t)(uintptr_t)(gt_cxys + (size_t)b * NMB_ * 2);
        const uint32_t lds = (uint32_t)(uintptr_t)(&s_cxy[0]);
        u32x4 g0;
        g0.x = 1u;                                        // count=1 (valid user D#)
        g0.y = lds;                                       // lds_addr
        g0.z = (uint32_t)(ga & 0xffffffffu);              // global_addr[31:0]
        g0.w = (uint32_t)((ga >> 32) & 0x01ffffffu)       // global_addr[56:32]
             | 0x80000000u;                               // type=2 ("image")
        i32x8 g1;
        g1[0] = (2 << 16);            // data_size=4B, workgroup_mask=0
        g1[1] = (int)(256u << 16);    // tensor_dim0[15:0] = 256 elements
        g1[2] = (int)(1u << 16);      // tensor_dim0 hi=0, tensor_dim1[15:0]=1
        g1[3] = (int)(256u << 16);    // tensor_dim1 hi=0, tile_dim0=256
        g1[4] = 1;                    // tile_dim1=1, tile_dim2=0
        g1[5] = 256;                  // tensor_dim0_stride lo
        g1[6] = 0;                    // stride hi / dim1_stride lo
        g1[7] = 0;
        i32x4 g2 = {0, 0, 0, 0};
        i32x4 g3 = {0, 0, 0, 0};
        i32x8 g4 = {0, 0, 0, 0, 0, 0, 0, 0};   // extra group, clang-23 6-arg form
        __builtin_amdgcn_tensor_load_to_lds(g0, g1, g2, g3, g4, 0);
        __builtin_amdgcn_s_wait_tensorcnt(0);
    }
#else
    if (threadIdx.x < NMB_ * 2)
        s_cxy[threadIdx.x] = gt_cxys[(size_t)b * NMB_ * 2 + threadIdx.x];
#endif
    if (threadIdx.x < NMB_) {
        const int n = threadIdx.x;
        s_m[n]         = (mask_gt[b * NMB_ + n] != 0.0f) ? 1 : 0;
        s_k[n * 3 + 0] = ws_topk[(b * NMB_ + n) * 3 + 0];
        s_k[n * 3 + 1] = ws_topk[(b * NMB_ + n) * 3 + 1];
        s_k[n * 3 + 2] = ws_topk[(b * NMB_ + n) * 3 + 2];
    }
    __syncthreads();

    const float gxc = (float)(g % GS_) + 0.5f;
    const float gyc = (float)(g / GS_) + 0.5f;

    float dmax = -INFINITY;
    int   nmax = 0, cnt = 0, first = -1;
    for (int n = 0; n < NMB_; ++n) {
        const float d = fabsf(gxc - s_cxy[2 * n]) + fabsf(gyc - s_cxy[2 * n + 1]);
        if (d > dmax) { dmax = d; nmax = n; }             // first-max tie-break (argmax)
        const bool mem = s_m[n] &&
            (g == s_k[n * 3] || g == s_k[n * 3 + 1] || g == s_k[n * 3 + 2]);
        if (mem) { cnt++; if (first < 0) first = n; }
    }

    int tgt, fg;
    if (cnt == 0)      { tgt = 0;     fg = 0; }
    else if (cnt == 1) { tgt = first; fg = 1; }
    else {
        const bool keep = s_m[nmax] &&
            (g == s_k[nmax * 3] || g == s_k[nmax * 3 + 1] || g == s_k[nmax * 3 + 2]);
        tgt = keep ? nmax : 0;
        fg  = keep ? 1 : 0;
    }
    const int cls = (int)gt_cls[b * NMB_ + tgt];
    ws_packed[b * NG_ + g] = tgt | (fg << 8) | (cls << 16);
}

// ---------------- K3a: boxes, fg mask, anc copy -----------------------------
__global__ __launch_bounds__(256)
void k_box_fg(const float* __restrict__ gt_cxys,
              const float* __restrict__ gt_whs,
              const float* __restrict__ anc_wh,
              const int*   __restrict__ ws_packed,
              float* __restrict__ out_box,
              float* __restrict__ out_fg,
              float* __restrict__ out_anc) {
    const int t = blockIdx.x * blockDim.x + threadIdx.x;   // (b,g)
    if (t == 0) {
#pragma unroll
        for (int i = 0; i < 6; ++i) out_anc[i] = anc_wh[i];
    }
    if (t >= BS_ * NG_) return;
    const int b = t / NG_, g = t % NG_;

    const int packed = ws_packed[t];
    const int tgt = packed & 0xff;
    const int fg  = (packed >> 8) & 1;

    const float cx = gt_cxys[(b * NMB_ + tgt) * 2 + 0];
    const float cy = gt_cxys[(b * NMB_ + tgt) * 2 + 1];
    const float w  = gt_whs [(b * NMB_ + tgt) * 2 + 0];
    const float h  = gt_whs [(b * NMB_ + tgt) * 2 + 1];
    const float tx = cx - (float)(g % GS_);
    const float ty = cy - (float)(g / GS_);
    const v4f box = {tx, ty, w, h};

#pragma unroll
    for (int a = 0; a < NA_; ++a) {
        __builtin_nontemporal_store(box,
            (v4f*)(out_box + ((size_t)(b * NA_ + a) * NG_ + g) * 4));
        const float aw = anc_wh[a * 2 + 0], ah = anc_wh[a * 2 + 1];
        const float rw = w / aw, rh = h / ah;
        const float m  = fmaxf(fmaxf(rw, 1.0f / rw), fmaxf(rh, 1.0f / rh));
        const float v  = (fg && (m < 4.0f)) ? 1.0f : 0.0f;
        __builtin_nontemporal_store(v, out_fg + (size_t)(b * NA_ + a) * NG_ + g);
    }
}

// ---------------- K3b: one-hot class scores (bandwidth-critical) ------------
// One thread per float4 of the (bs, NA, ng, 80) tensor; NT b128 stores.
__global__ __launch_bounds__(256)
void k_scores(const int* __restrict__ ws_packed, float* __restrict__ out_scores) {
    const int t = blockIdx.x * blockDim.x + threadIdx.x;   // quad index
    const int q = t % 20;                                  // class quad [0,20)
    int r = t / 20;
    const int g = r % NG_;
    r /= NG_;                                              // r = b*NA + a
    const int b = r / NA_;

    const int cls  = (ws_packed[b * NG_ + g] >> 16) & 0xff;
    const int base = q * 4;
    v4f v;
    v.x = (cls == base + 0) ? 1.0f : 0.0f;
    v.y = (cls == base + 1) ? 1.0f : 0.0f;
    v.z = (cls == base + 2) ? 1.0f : 0.0f;
    v.w = (cls == base + 3) ? 1.0f : 0.0f;
    __builtin_nontemporal_store(v, (v4f*)(out_scores + (size_t)t * 4));
}

// ---------------- host-side launch ------------------------------------------
extern "C" void kernel_launch(void* const* d_in, const int* in_sizes, int n_in,
                              void* d_out, int out_size, void* d_ws, size_t ws_size,
                              hipStream_t stream) {
    (void)in_sizes; (void)n_in; (void)out_size; (void)ws_size;
    const float* anc_wh  = (const float*)d_in[0];
    const float* grid_in = (const float*)d_in[1]; (void)grid_in; // analytic (bit-exact)
    const float* gt_cls  = (const float*)d_in[2];
    const float* gt_cxys = (const float*)d_in[3];
    const float* gt_whs  = (const float*)d_in[4];
    const float* mask_gt = (const float*)d_in[5];

    float* out        = (float*)d_out;
    float* out_box    = out;                                           // 32*3*6400*4
    float* out_scores = out + (size_t)BS_ * NA_ * NG_ * 4;             // 32*3*6400*80
    float* out_anc    = out_scores + (size_t)BS_ * NA_ * NG_ * NCLS_;  // 6
    float* out_fg     = out_anc + 6;                                   // 32*3*6400

    int* ws_topk   = (int*)d_ws;                  // 4096*3 ints  (48 KB)
    int* ws_packed = ws_topk + BS_ * NMB_ * 3;    // 32*6400 ints (800 KB)

    k_topk  <<<(BS_ * NMB_) / 8,        256, 0, stream>>>(gt_cxys, ws_topk);
    k_assign<<<BS_ * (NG_ / 256),       256, 0, stream>>>(gt_cxys, gt_cls, mask_gt,
                                                          ws_topk, ws_packed);
    k_box_fg<<<(BS_ * NG_) / 256,       256, 0, stream>>>(gt_cxys, gt_whs, anc_wh,
                                                          ws_packed, out_box, out_fg, out_anc);
    k_scores<<<(BS_ * NA_ * NG_ * 20) / 256, 256, 0, stream>>>(ws_packed, out_scores);
}